// SimpleDiffNDCGLoss_22917945491556
// MI455X (gfx1250) — compile-verified
//
#include <hip/hip_runtime.h>
#include <hip/hip_bf16.h>
#include <stdint.h>

#define TPB 256
#define TILE_ROWS 256
#define TILE_FLOATS (TILE_ROWS * 8)   // 2048 floats = 8 KB per tensor per tile
#define TILE_BYTES (TILE_FLOATS * 4)

// ---- CDNA5 async global->LDS copy (ASYNCcnt-tracked), inline asm ----
// GVS addressing: saddr(64b SGPR pair) + vaddr(32b byte offset). VDST holds
// the LDS byte address (low 32 bits of the generic shared pointer).
__device__ __forceinline__ void async_ld_b128(uint32_t lds_byte_addr,
                                              const void* gbase,
                                              uint32_t gbyte_off) {
  asm volatile("global_load_async_to_lds_b128 %0, %1, %2"
               :
               : "v"(lds_byte_addr), "v"(gbyte_off), "s"(gbase)
               : "memory");
}
__device__ __forceinline__ void wait_async_le4() {
  asm volatile("s_wait_asynccnt 0x4" ::: "memory");
}
__device__ __forceinline__ void wait_async_0() {
  asm volatile("s_wait_asynccnt 0x0" ::: "memory");
}

// ---- per-row math: soft odd-even sort of -scores, labels pushed through the
// same soft-swaps (y == labels @ P identity), then DCG / ideal DCG ----
__device__ __forceinline__ float row_loss(const float4 s0, const float4 s1,
                                          const float4 l0, const float4 l1) {
  float x[8] = {-s0.x, -s0.y, -s0.z, -s0.w, -s1.x, -s1.y, -s1.z, -s1.w};
  float y[8] = {l0.x, l0.y, l0.z, l0.w, l1.x, l1.y, l1.z, l1.w};
  float s[8] = {l0.x, l0.y, l0.z, l0.w, l1.x, l1.y, l1.z, l1.w};

#define SOFT_SWAP(i, j)                                                     \
  {                                                                         \
    float a_ = x[i], b_ = x[j];                                             \
    float al_ = 0.31830988618379067f * atanf(10.0f * (b_ - a_)) + 0.5f;     \
    float be_ = 1.0f - al_;                                                 \
    x[i] = al_ * a_ + be_ * b_;                                             \
    x[j] = al_ * b_ + be_ * a_;                                             \
    float p_ = y[i], q_ = y[j];                                             \
    y[i] = al_ * p_ + be_ * q_;                                             \
    y[j] = al_ * q_ + be_ * p_;                                             \
  }
#define EVEN_L SOFT_SWAP(0, 1) SOFT_SWAP(2, 3) SOFT_SWAP(4, 5) SOFT_SWAP(6, 7)
#define ODD_L  SOFT_SWAP(1, 2) SOFT_SWAP(3, 4) SOFT_SWAP(5, 6)
  EVEN_L ODD_L EVEN_L ODD_L EVEN_L ODD_L EVEN_L ODD_L
#undef EVEN_L
#undef ODD_L
#undef SOFT_SWAP

  // discounts 1/log2(k+2)
  const float D0 = 1.0f;
  const float D1 = 0.6309297535714575f;
  const float D2 = 0.5f;
  const float D3 = 0.43067655807339306f;
  const float D4 = 0.38685280723454163f;
  const float D5 = 0.3562071871080222f;
  const float D6 = 0.3333333333333333f;
  const float D7 = 0.31546487678572877f;

  float dcg = (exp2f(y[0]) - 1.0f) * D0 + (exp2f(y[1]) - 1.0f) * D1 +
              (exp2f(y[2]) - 1.0f) * D2 + (exp2f(y[3]) - 1.0f) * D3 +
              (exp2f(y[4]) - 1.0f) * D4 + (exp2f(y[5]) - 1.0f) * D5 +
              (exp2f(y[6]) - 1.0f) * D6 + (exp2f(y[7]) - 1.0f) * D7;

  // hard descending sort of labels: optimal 19-comparator network for n=8
#define CSWAP(i, j)                      \
  {                                      \
    float hi_ = fmaxf(s[i], s[j]);       \
    float lo_ = fminf(s[i], s[j]);       \
    s[i] = hi_;                          \
    s[j] = lo_;                          \
  }
  CSWAP(0, 1) CSWAP(2, 3) CSWAP(4, 5) CSWAP(6, 7)
  CSWAP(0, 2) CSWAP(1, 3) CSWAP(4, 6) CSWAP(5, 7)
  CSWAP(1, 2) CSWAP(5, 6) CSWAP(0, 4) CSWAP(3, 7)
  CSWAP(1, 5) CSWAP(2, 6)
  CSWAP(1, 4) CSWAP(3, 6)
  CSWAP(2, 4) CSWAP(3, 5)
  CSWAP(3, 4)
#undef CSWAP

  float idcg = (exp2f(s[0]) - 1.0f) * D0 + (exp2f(s[1]) - 1.0f) * D1 +
               (exp2f(s[2]) - 1.0f) * D2 + (exp2f(s[3]) - 1.0f) * D3 +
               (exp2f(s[4]) - 1.0f) * D4 + (exp2f(s[5]) - 1.0f) * D5 +
               (exp2f(s[6]) - 1.0f) * D6 + (exp2f(s[7]) - 1.0f) * D7;

  return 1.0f - dcg / fmaxf(idcg, 1e-10f);
}

// One thread = one row. Each wave async-copies its own 32 rows (1 KB scores +
// 1 KB labels) to a private LDS slice; double-buffered across tiles; no
// barriers needed since slices are wave-local and waits are per-wave ASYNCcnt.
__global__ __launch_bounds__(TPB) void diff_ndcg_kernel(
    const float* __restrict__ scores, const float* __restrict__ labels,
    float* __restrict__ out, int B, int fullTiles) {
  __shared__ __align__(16) float sS[2][TILE_FLOATS];
  __shared__ __align__(16) float sL[2][TILE_FLOATS];

  const int tid = (int)threadIdx.x;
  const int w = tid >> 5;   // wave id (wave32)
  const int ln = tid & 31;  // lane id
  const int G = (int)gridDim.x;
  // each wave owns bytes [w*1024, w*1024+1024) of each tile buffer; each lane
  // copies two 16B chunks per tensor: off0 and off0+512
  const uint32_t off0 = (uint32_t)(w * 1024 + ln * 16);

  int t = (int)blockIdx.x;
  int buf = 0;

  if (t < fullTiles) {  // prologue: issue tile t into buffer 0
    const char* gS = (const char*)scores + (size_t)t * TILE_BYTES;
    const char* gL = (const char*)labels + (size_t)t * TILE_BYTES;
    uint32_t dS = (uint32_t)(uintptr_t)((const char*)&sS[0][0]) + off0;
    uint32_t dL = (uint32_t)(uintptr_t)((const char*)&sL[0][0]) + off0;
    async_ld_b128(dS, gS, off0);
    async_ld_b128(dS + 512, gS, off0 + 512);
    async_ld_b128(dL, gL, off0);
    async_ld_b128(dL + 512, gL, off0 + 512);
  }

  while (t < fullTiles) {
    const int tn = t + G;
    if (tn < fullTiles) {  // prefetch next tile into the other buffer
      const int nb = buf ^ 1;
      const char* gS = (const char*)scores + (size_t)tn * TILE_BYTES;
      const char* gL = (const char*)labels + (size_t)tn * TILE_BYTES;
      uint32_t dS = (uint32_t)(uintptr_t)((const char*)&sS[nb][0]) + off0;
      uint32_t dL = (uint32_t)(uintptr_t)((const char*)&sL[nb][0]) + off0;
      async_ld_b128(dS, gS, off0);
      async_ld_b128(dS + 512, gS, off0 + 512);
      async_ld_b128(dL, gL, off0);
      async_ld_b128(dL + 512, gL, off0 + 512);
      wait_async_le4();  // async loads complete in order: current tile done
    } else {
      wait_async_0();
    }

    const float4* s4 = reinterpret_cast<const float4*>(&sS[buf][tid * 8]);
    const float4* l4 = reinterpret_cast<const float4*>(&sL[buf][tid * 8]);
    float4 a0 = s4[0], a1 = s4[1];
    float4 b0 = l4[0], b1 = l4[1];
    out[(size_t)t * TILE_ROWS + tid] = row_loss(a0, a1, b0, b1);

    buf ^= 1;
    t = tn;
  }

  // tail rows (B not multiple of 256): direct global loads, block 0 only
  if (blockIdx.x == 0) {
    int row = fullTiles * TILE_ROWS + tid;
    if (row < B) {
      const float4* s4 =
          reinterpret_cast<const float4*>(scores + (size_t)row * 8);
      const float4* l4 =
          reinterpret_cast<const float4*>(labels + (size_t)row * 8);
      out[row] = row_loss(s4[0], s4[1], l4[0], l4[1]);
    }
  }
}

extern "C" void kernel_launch(void* const* d_in, const int* in_sizes, int n_in,
                              void* d_out, int out_size, void* d_ws,
                              size_t ws_size, hipStream_t stream) {
  (void)n_in;
  (void)out_size;
  (void)d_ws;
  (void)ws_size;
  const float* scores = (const float*)d_in[0];
  const float* labels = (const float*)d_in[1];
  float* out = (float*)d_out;

  const int B = in_sizes[0] / 8;
  const int fullTiles = B / TILE_ROWS;
  int grid = fullTiles;
  if (grid > 512) grid = 512;  // >=2 tiles/block -> async double-buffer overlap
  if (grid < 1) grid = 1;

  diff_ndcg_kernel<<<grid, TPB, 0, stream>>>(scores, labels, out, B, fullTiles);
}